// BEVFormerEncoderLayer_53747220742148
// MI455X (gfx1250) — compile-verified
//
#include <hip/hip_runtime.h>
#include <hip/hip_bf16.h>

typedef float v2f __attribute__((ext_vector_type(2)));
typedef float v8f __attribute__((ext_vector_type(8)));

#define N_Q 10000
#define D_M 256
#define MAX_K 1024

// ---------------------------------------------------------------------------
// LayerNorm over last dim (256). One wave per row, 8 rows per 256-thread block.
// ---------------------------------------------------------------------------
__global__ void ln_kernel(const float* __restrict__ x, const float* __restrict__ g,
                          const float* __restrict__ bta, float* __restrict__ y, int nrows) {
  int row = blockIdx.x * 8 + (threadIdx.x >> 5);
  if (row >= nrows) return;
  int lane = threadIdx.x & 31;
  const float4* xr = (const float4*)(x + (size_t)row * D_M);
  float4 a = xr[lane];
  float4 b = xr[lane + 32];
  float s = a.x + a.y + a.z + a.w + b.x + b.y + b.z + b.w;
#pragma unroll
  for (int o = 16; o > 0; o >>= 1) s += __shfl_xor(s, o, 32);
  float mu = s * (1.0f / 256.0f);
  float d, ss = 0.f;
  d = a.x - mu; ss += d * d;  d = a.y - mu; ss += d * d;
  d = a.z - mu; ss += d * d;  d = a.w - mu; ss += d * d;
  d = b.x - mu; ss += d * d;  d = b.y - mu; ss += d * d;
  d = b.z - mu; ss += d * d;  d = b.w - mu; ss += d * d;
#pragma unroll
  for (int o = 16; o > 0; o >>= 1) ss += __shfl_xor(ss, o, 32);
  float inv = rsqrtf(ss * (1.0f / 256.0f) + 1e-5f);
  const float4* g4 = (const float4*)g;
  const float4* b4 = (const float4*)bta;
  float4* yr = (float4*)(y + (size_t)row * D_M);
  float4 gv = g4[lane], bv = b4[lane], o0;
  o0.x = (a.x - mu) * inv * gv.x + bv.x;
  o0.y = (a.y - mu) * inv * gv.y + bv.y;
  o0.z = (a.z - mu) * inv * gv.z + bv.z;
  o0.w = (a.w - mu) * inv * gv.w + bv.w;
  yr[lane] = o0;
  gv = g4[lane + 32]; bv = b4[lane + 32];
  o0.x = (b.x - mu) * inv * gv.x + bv.x;
  o0.y = (b.y - mu) * inv * gv.y + bv.y;
  o0.z = (b.z - mu) * inv * gv.z + bv.z;
  o0.w = (b.w - mu) * inv * gv.w + bv.w;
  yr[lane + 32] = o0;
}

// ---------------------------------------------------------------------------
// WMMA f32 GEMM: C[M x ?] = concat(A0,A1)[M x (K0+K1)] @ W[(K0+K1) x ldw]
// (+ bias, + optional residual (stride ldc), + optional relu).
// Block = 8 waves, one 16-wide N tile per block. The K x 16 weight panel is
// staged once per block into LDS with GLOBAL_LOAD_ASYNC_TO_LDS_B128
// (ASYNCcnt -> s_wait_asynccnt), then each wave runs V_WMMA_F32_16X16X4_F32
// on its own 16-row M tile reading B fragments from LDS (bank-conflict free).
// ---------------------------------------------------------------------------
__global__ void gemm_wmma(const float* __restrict__ A0, const float* __restrict__ A1,
                          int K0, int K1,
                          const float* __restrict__ W, int ldw,
                          const float* __restrict__ bias,
                          const float* __restrict__ resid,
                          float* __restrict__ C, int ldc,
                          int M, int act) {
  __shared__ __align__(16) float bpanel[MAX_K * 16];
  int tid = threadIdx.x;
  int wave = tid >> 5;
  int lane = tid & 31;
  int nb = blockIdx.x * 16;
  int K = K0 + K1;

  // ---- async-stage the B panel (K rows x 16 cols) into LDS ----
  // K*16 is always a multiple of 1024 here -> uniform trip count, EXEC all-1s.
  {
    unsigned ldsbase = (unsigned)(size_t)(&bpanel[0]);
    for (int p = tid * 4; p < K * 16; p += 1024) {
      int row = p >> 4, col = p & 15;
      unsigned voff = (unsigned)((row * ldw + nb + col) * 4);  // byte offset
      unsigned dst = ldsbase + (unsigned)(p * 4);
      asm volatile("global_load_async_to_lds_b128 %0, %1, %2 offset:0"
                   :: "v"(dst), "v"(voff), "s"(W)
                   : "memory");
    }
    asm volatile("s_wait_asynccnt 0x0" ::: "memory");
  }
  __syncthreads();

  int mt = blockIdx.y * 8 + wave;
  if (mt * 16 >= M) return;                 // wave-uniform: EXEC all-1s below
  int mbase = mt * 16;
  int half = lane >> 4;
  int l16 = lane & 15;
  size_t r = (size_t)(mbase + l16);
  const float* bp = &bpanel[half * 32 + l16];   // rows (k0+2*half), (k0+2*half+1)
  v8f acc = {};

  const float* ap = A0 + r * (size_t)K0 + 2 * half;
#pragma unroll 4
  for (int k0 = 0; k0 < K0; k0 += 4) {
    __builtin_prefetch(ap + 32, 0, 3);
    v2f a = *(const v2f*)ap;                // A frag: V0={K0|K2}, V1={K1|K3}
    ap += 4;
    v2f b;
    b.x = bp[k0 * 16];
    b.y = bp[k0 * 16 + 16];
    acc = __builtin_amdgcn_wmma_f32_16x16x4_f32(
        false, a, false, b, (short)0, acc, false, false);
  }
  if (A1) {
    ap = A1 + r * (size_t)K1 + 2 * half;
#pragma unroll 4
    for (int k0 = K0; k0 < K; k0 += 4) {
      __builtin_prefetch(ap + 32, 0, 3);
      v2f a = *(const v2f*)ap;
      ap += 4;
      v2f b;
      b.x = bp[k0 * 16];
      b.y = bp[k0 * 16 + 16];
      acc = __builtin_amdgcn_wmma_f32_16x16x4_f32(
          false, a, false, b, (short)0, acc, false, false);
    }
  }

  float bv = bias ? bias[nb + l16] : 0.f;
#pragma unroll
  for (int rr = 0; rr < 8; rr++) {          // C: VGPR rr -> rows rr / rr+8
    int m = mbase + rr + 8 * half;
    float v = acc[rr] + bv;
    if (resid) v += resid[(size_t)m * ldc + nb + l16];
    if (act) v = fmaxf(v, 0.f);
    C[(size_t)m * ldc + nb + l16] = v;
  }
}

// ---------------------------------------------------------------------------
// Bilinear sample: src indexed as src[(y*W + x) * pixstride], OOB -> 0.
// ---------------------------------------------------------------------------
__device__ __forceinline__ float bilin(const float* __restrict__ src, float ix, float iy,
                                       int H, int W, int pixstride) {
  float x0f = floorf(ix), y0f = floorf(iy);
  int x0 = (int)x0f, y0 = (int)y0f;
  float wx1 = ix - x0f, wy1 = iy - y0f;
  float wx0 = 1.f - wx1, wy0 = 1.f - wy1;
  float v = 0.f;
  bool xin0 = (x0 >= 0) & (x0 < W);
  bool xin1 = (x0 + 1 >= 0) & (x0 + 1 < W);
  if (y0 >= 0 && y0 < H) {
    if (xin0) v += wx0 * wy0 * src[((size_t)y0 * W + x0) * pixstride];
    if (xin1) v += wx1 * wy0 * src[((size_t)y0 * W + x0 + 1) * pixstride];
  }
  if (y0 + 1 >= 0 && y0 + 1 < H) {
    if (xin0) v += wx0 * wy1 * src[((size_t)(y0 + 1) * W + x0) * pixstride];
    if (xin1) v += wx1 * wy1 * src[((size_t)(y0 + 1) * W + x0 + 1) * pixstride];
  }
  return v;
}

// ---------------------------------------------------------------------------
// Temporal sampling. One block (256 thr) per query: thread = (head h, chan c).
// raw = [toff(128) | twl(64)] pre-activation. BEV maps are pb/q1 row-major
// (pixel-major, channel minor) -> coalesced 32-channel gathers per corner.
// ---------------------------------------------------------------------------
__global__ void t_sample_kernel(const float* __restrict__ raw, const float* __restrict__ ref2d,
                                const float* __restrict__ ego, const float* __restrict__ pb,
                                const float* __restrict__ q1, float* __restrict__ tf) {
  __shared__ float sx[2][8][4], sy[2][8][4], sw[2][8][4];
  int n = blockIdx.x;
  int tid = threadIdx.x;
  const float* r = raw + (size_t)n * 192;
  if (tid < 64) {
    int si = tid >> 5, hh = (tid >> 2) & 7, p = tid & 3;
    float ox = tanhf(r[((si * 8 + hh) * 4 + p) * 2 + 0]) * 0.15f;
    float oy = tanhf(r[((si * 8 + hh) * 4 + p) * 2 + 1]) * 0.15f;
    float bx = ref2d[2 * n + 0] + (si == 0 ? ego[0] : 0.f);
    float by = ref2d[2 * n + 1] + (si == 0 ? ego[1] : 0.f);
    sx[si][hh][p] = (bx + ox) * 100.f - 0.5f;
    sy[si][hh][p] = (by + oy) * 100.f - 0.5f;
  } else if (tid < 80) {
    int idx = tid - 64, si = idx >> 3, hh = idx & 7;
    const float* l = r + 128 + (si * 8 + hh) * 4;
    float l0 = l[0], l1 = l[1], l2 = l[2], l3 = l[3];
    float mx = fmaxf(fmaxf(l0, l1), fmaxf(l2, l3));
    float e0 = expf(l0 - mx), e1 = expf(l1 - mx), e2 = expf(l2 - mx), e3 = expf(l3 - mx);
    float inv = 1.f / (e0 + e1 + e2 + e3);
    sw[si][hh][0] = e0 * inv; sw[si][hh][1] = e1 * inv;
    sw[si][hh][2] = e2 * inv; sw[si][hh][3] = e3 * inv;
  }
  __syncthreads();
  int c = tid & 31, h = tid >> 5;
  const float* s0 = pb + h * 32 + c;   // si=0: prev map
  const float* s1 = q1 + h * 32 + c;   // si=1: cur map
  float acc = 0.f;
#pragma unroll
  for (int p = 0; p < 4; p++)
    acc += sw[0][h][p] * bilin(s0, sx[0][h][p], sy[0][h][p], 100, 100, D_M);
#pragma unroll
  for (int p = 0; p < 4; p++)
    acc += sw[1][h][p] * bilin(s1, sx[1][h][p], sy[1][h][p], 100, 100, D_M);
  tf[(size_t)n * D_M + h * 32 + c] = acc * 0.5f;
}

// ---------------------------------------------------------------------------
// Spatial sampling over 6 cameras with masked softmax.
// raw = [soff(256) | swl(128)] pre-activation. image_feats channel-major.
// ---------------------------------------------------------------------------
__global__ void s_sample_kernel(const float* __restrict__ raw, const float* __restrict__ refcam,
                                const unsigned char* __restrict__ bmask,
                                const float* __restrict__ imf, float* __restrict__ sf) {
  __shared__ float px[8][16], py[8][16], wgt[8][16];
  __shared__ float srefx[4], srefy[4];
  __shared__ int svis[4];
  int n = blockIdx.x, tid = threadIdx.x;
  int c = tid & 31, h = tid >> 5;
  const float* r = raw + (size_t)n * 384;
  float acc = 0.f, cnt = 0.f;
  for (int cam = 0; cam < 6; cam++) {
    __syncthreads();
    if (tid < 4) {
      size_t base = (size_t)(cam * N_Q + n) * 4 + tid;
      srefx[tid] = refcam[base * 2 + 0];
      srefy[tid] = refcam[base * 2 + 1];
      svis[tid] = (int)bmask[base];
    }
    __syncthreads();
    if (tid < 128) {
      int hh = tid >> 4, pt = tid & 15, pp = pt >> 2;
      float ox = tanhf(r[(hh * 16 + pt) * 2 + 0]) * 0.12f;
      float oy = tanhf(r[(hh * 16 + pt) * 2 + 1]) * 0.12f;
      px[hh][pt] = (srefx[pp] + ox) * 60.f - 0.5f;
      py[hh][pt] = (srefy[pp] + oy) * 24.f - 0.5f;
    } else if (tid < 136) {
      int hh = tid - 128;
      const float* l = r + 256 + hh * 16;
      float mx = -1e30f;
      for (int k = 0; k < 16; k++)
        if (svis[k >> 2]) mx = fmaxf(mx, l[k]);
      float e[16], ssum = 0.f;
      for (int k = 0; k < 16; k++) {
        float v = svis[k >> 2] ? expf(l[k] - mx) : 0.f;
        e[k] = v; ssum += v;
      }
      float inv = (ssum > 0.f) ? 1.f / ssum : 0.f;
      for (int k = 0; k < 16; k++) wgt[hh][k] = e[k] * inv;
    }
    __syncthreads();
    int anyv = svis[0] | svis[1] | svis[2] | svis[3];
    if (anyv) {
      const float* src = imf + ((size_t)cam * 256 + h * 32 + c) * (24 * 60);
      float a = 0.f;
      for (int pt = 0; pt < 16; pt++) {
        float w = wgt[h][pt];
        if (w != 0.f) a += w * bilin(src, px[h][pt], py[h][pt], 24, 60, 1);
      }
      acc += a;
      cnt += 1.f;
    }
  }
  sf[(size_t)n * D_M + h * 32 + c] = acc / fmaxf(cnt, 1.f);
}

// ---------------------------------------------------------------------------
extern "C" void kernel_launch(void* const* d_in, const int* in_sizes, int n_in,
                              void* d_out, int out_size, void* d_ws, size_t ws_size,
                              hipStream_t stream) {
  const float* query  = (const float*)d_in[0];
  const float* prevb  = (const float*)d_in[1];
  const float* imf    = (const float*)d_in[2];
  const float* ref2d  = (const float*)d_in[3];
  const float* refcam = (const float*)d_in[4];
  const float* ego    = (const float*)d_in[5];
  const float* t_off_w = (const float*)d_in[6];
  const float* t_off_b = (const float*)d_in[7];
  const float* t_wt_w  = (const float*)d_in[8];
  const float* t_wt_b  = (const float*)d_in[9];
  const float* t_out_w = (const float*)d_in[10];
  const float* t_out_b = (const float*)d_in[11];
  const float* s_off_w = (const float*)d_in[12];
  const float* s_off_b = (const float*)d_in[13];
  const float* s_wt_w  = (const float*)d_in[14];
  const float* s_wt_b  = (const float*)d_in[15];
  const float* s_out_w = (const float*)d_in[16];
  const float* s_out_b = (const float*)d_in[17];
  const float* ffn_w1  = (const float*)d_in[18];
  const float* ffn_b1  = (const float*)d_in[19];
  const float* ffn_w2  = (const float*)d_in[20];
  const float* ffn_b2  = (const float*)d_in[21];
  const float* ln1_g = (const float*)d_in[22];
  const float* ln1_b = (const float*)d_in[23];
  const float* ln2_g = (const float*)d_in[24];
  const float* ln2_b = (const float*)d_in[25];
  const float* ln3_g = (const float*)d_in[26];
  const float* ln3_b = (const float*)d_in[27];
  const unsigned char* bmask = (const unsigned char*)d_in[28];

  float* ws  = (float*)d_ws;
  float* q1  = ws;                 // N*256  : q1, later q2
  float* pb  = ws + 2560000;       // N*256  : pb, later s_fused
  float* xb  = ws + 5120000;       // N*256  : x (after temporal), later ln3(x2)
  float* x2  = ws + 7680000;       // N*256  : x after spatial
  float* raw = ws + 10240000;      // N*384  : raw192 then raw384
  float* tf  = ws + 14080000;      // N*1024 : t_fused, later FFN hidden
  float* out = (float*)d_out;

  dim3 blk(256);
  const int GY = 79;  // ceil(625 m-tiles / 8 waves)

  // q1 = LN(query), pb = LN(prev_bev)
  ln_kernel<<<1250, blk, 0, stream>>>(query, ln1_g, ln1_b, q1, N_Q);
  ln_kernel<<<1250, blk, 0, stream>>>(prevb, ln1_g, ln1_b, pb, N_Q);
  // raw192 = concat(pb,q1) @ [t_off_w | t_wt_w]
  gemm_wmma<<<dim3(8, GY), blk, 0, stream>>>(pb, q1, 256, 256, t_off_w, 128, t_off_b,
                                             nullptr, raw, 192, N_Q, 0);
  gemm_wmma<<<dim3(4, GY), blk, 0, stream>>>(pb, q1, 256, 256, t_wt_w, 64, t_wt_b,
                                             nullptr, raw + 128, 192, N_Q, 0);
  // temporal deformable sampling -> t_fused
  t_sample_kernel<<<N_Q, blk, 0, stream>>>(raw, ref2d, ego, pb, q1, tf);
  // x = query + t_fused @ t_out_w + b
  gemm_wmma<<<dim3(16, GY), blk, 0, stream>>>(tf, nullptr, 256, 0, t_out_w, 256, t_out_b,
                                              query, xb, 256, N_Q, 0);
  // q2 = LN2(x)  (reuse q1 buffer)
  ln_kernel<<<1250, blk, 0, stream>>>(xb, ln2_g, ln2_b, q1, N_Q);
  // raw384 = q2 @ [s_off_w | s_wt_w]
  gemm_wmma<<<dim3(16, GY), blk, 0, stream>>>(q1, nullptr, 256, 0, s_off_w, 256, s_off_b,
                                              nullptr, raw, 384, N_Q, 0);
  gemm_wmma<<<dim3(8, GY), blk, 0, stream>>>(q1, nullptr, 256, 0, s_wt_w, 128, s_wt_b,
                                             nullptr, raw + 256, 384, N_Q, 0);
  // spatial deformable sampling -> s_fused (reuse pb buffer)
  s_sample_kernel<<<N_Q, blk, 0, stream>>>(raw, refcam, bmask, imf, pb);
  // x2 = x + s_fused @ s_out_w + b
  gemm_wmma<<<dim3(16, GY), blk, 0, stream>>>(pb, nullptr, 256, 0, s_out_w, 256, s_out_b,
                                              xb, x2, 256, N_Q, 0);
  // ln3 = LN3(x2)  (reuse xb buffer)
  ln_kernel<<<1250, blk, 0, stream>>>(x2, ln3_g, ln3_b, xb, N_Q);
  // h = relu(ln3 @ ffn_w1 + b1)  (reuse tf buffer, N*1024)
  gemm_wmma<<<dim3(64, GY), blk, 0, stream>>>(xb, nullptr, 256, 0, ffn_w1, 1024, ffn_b1,
                                              nullptr, tf, 1024, N_Q, 1);
  // out = x2 + h @ ffn_w2 + b2
  gemm_wmma<<<dim3(16, GY), blk, 0, stream>>>(tf, nullptr, 1024, 0, ffn_w2, 256, ffn_b2,
                                              x2, out, 256, N_Q, 0);
}